// Model_15058155340185
// MI455X (gfx1250) — compile-verified
//
#include <hip/hip_runtime.h>
#include <hip/hip_bf16.h>

typedef float v2f __attribute__((ext_vector_type(2)));
typedef float v8f __attribute__((ext_vector_type(8)));

#define FEATS 128

// One wave (32 lanes) per edge: lane loads float4 of the 512B source row
// (perfectly coalesced), then 4 f32 atomics into agg[dst] (L2-resident).
__global__ __launch_bounds__(256) void sage_scatter(
    const float* __restrict__ x, const int* __restrict__ src,
    const int* __restrict__ dst, float* __restrict__ agg,
    float* __restrict__ deg, int nedges) {
  int wave = (int)((blockIdx.x * blockDim.x + threadIdx.x) >> 5);
  int lane = threadIdx.x & 31;
  if (wave >= nedges) return;
  int s = src[wave];
  int d = dst[wave];
  const float4* xrow = (const float4*)(x + (size_t)s * FEATS);
  float4 v = xrow[lane];
  float* arow = agg + (size_t)d * FEATS + lane * 4;
  atomicAdd(arow + 0, v.x);
  atomicAdd(arow + 1, v.y);
  atomicAdd(arow + 2, v.z);
  atomicAdd(arow + 3, v.w);
  if (lane == 0) atomicAdd(deg + d, 1.0f);
}

// out[0:ndst] = act( hself[0:ndst] @ Wself + (agg/max(deg,1)) @ Wneigh + bias )
// Block: 256 threads = 8 waves. Block owns a 16-row slab; wave w owns columns
// [16w, 16w+16). K=128 swept in steps of 4 with V_WMMA_F32_16X16X4_F32,
// two WMMAs (self + neigh) per step into one f32 accumulator.
__global__ __launch_bounds__(256) void sage_gemm(
    const float* __restrict__ hself, const float* __restrict__ agg,
    const float* __restrict__ deg, const float* __restrict__ Wself,
    const float* __restrict__ Wneigh, const float* __restrict__ bias,
    float* __restrict__ out, int ndst, int do_relu) {
  extern __shared__ float lds[];
  float* sWs = lds;                       // 128*128
  float* sWn = lds + FEATS * FEATS;       // 128*128
  float* sB  = lds + 2 * FEATS * FEATS;   // 128
  for (int i = threadIdx.x; i < FEATS * FEATS; i += blockDim.x) {
    sWs[i] = Wself[i];
    sWn[i] = Wneigh[i];
  }
  if (threadIdx.x < FEATS) sB[threadIdx.x] = bias[threadIdx.x];
  __syncthreads();

  int rowbase = blockIdx.x * 16;
  if (rowbase >= ndst) return;

  int lane  = threadIdx.x & 31;
  int waveN = (threadIdx.x >> 5);     // 0..7 -> 16-col tile
  int m     = lane & 15;              // A-matrix row / D-matrix col lane id
  int khalf = (lane >> 4) * 2;        // K sub-offset per half-wave: 0 or 2
  int ncol  = waveN * 16 + m;         // B/D column for this lane
  int row   = rowbase + m;            // A row for this lane

  float rdeg = 1.0f / fmaxf(deg[row], 1.0f);
  const float* aSelf = hself + (size_t)row * FEATS;
  const float* aAgg  = agg   + (size_t)row * FEATS;

  v8f c = {0.f, 0.f, 0.f, 0.f, 0.f, 0.f, 0.f, 0.f};
  #pragma unroll 4
  for (int k = 0; k < FEATS; k += 4) {
    int kk = k + khalf;
    v2f as, an, bs, bn;
    as[0] = aSelf[kk];
    as[1] = aSelf[kk + 1];
    an[0] = aAgg[kk] * rdeg;
    an[1] = aAgg[kk + 1] * rdeg;
    bs[0] = sWs[kk * FEATS + ncol];
    bs[1] = sWs[(kk + 1) * FEATS + ncol];
    bn[0] = sWn[kk * FEATS + ncol];
    bn[1] = sWn[(kk + 1) * FEATS + ncol];
    // D = A*B + C, full fp32 matrix pipe (V_WMMA_F32_16X16X4_F32)
    c = __builtin_amdgcn_wmma_f32_16x16x4_f32(false, as, false, bs,
                                              (short)0, c, false, false);
    c = __builtin_amdgcn_wmma_f32_16x16x4_f32(false, an, false, bn,
                                              (short)0, c, false, false);
  }

  float bcol = sB[ncol];
  int rhalf = (lane >> 4) * 8;   // D: vgpr i holds row i (lanes 0-15) / i+8 (16-31)
  #pragma unroll
  for (int i = 0; i < 8; i++) {
    float v = c[i] + bcol;
    if (do_relu) v = fmaxf(v, 0.0f);
    out[(size_t)(rowbase + rhalf + i) * FEATS + ncol] = v;
  }
}

extern "C" void kernel_launch(void* const* d_in, const int* in_sizes, int n_in,
                              void* d_out, int out_size, void* d_ws, size_t ws_size,
                              hipStream_t stream) {
  (void)n_in; (void)out_size; (void)ws_size;
  const float* x       = (const float*)d_in[0];
  const int*   src1    = (const int*)d_in[1];
  const int*   dst1    = (const int*)d_in[2];
  const int*   src2    = (const int*)d_in[3];
  const int*   dst2    = (const int*)d_in[4];
  const float* Wself1  = (const float*)d_in[7];
  const float* Wneigh1 = (const float*)d_in[8];
  const float* b1      = (const float*)d_in[9];
  const float* Wself2  = (const float*)d_in[10];
  const float* Wneigh2 = (const float*)d_in[11];
  const float* b2      = (const float*)d_in[12];

  const int E1  = in_sizes[1];
  const int E2  = in_sizes[3];
  const int ND1 = 61440;
  const int ND2 = 4096;

  // Workspace layout (agg/deg contiguous per layer so one memset covers both)
  char*  ws   = (char*)d_ws;
  size_t off  = 0;
  float* agg1 = (float*)(ws + off); off += (size_t)ND1 * FEATS * sizeof(float);
  float* deg1 = (float*)(ws + off); off += (size_t)ND1 * sizeof(float);
  float* h1   = (float*)(ws + off); off += (size_t)ND1 * FEATS * sizeof(float);
  float* agg2 = (float*)(ws + off); off += (size_t)ND2 * FEATS * sizeof(float);
  float* deg2 = (float*)(ws + off); off += (size_t)ND2 * sizeof(float);

  dim3 blk(256);
  const int wpb = 8;  // waves (edges) per block
  size_t shmem = (size_t)(2 * FEATS * FEATS + FEATS) * sizeof(float); // ~128.5KB of 320KB LDS

  // ---- Layer 1 ----
  hipMemsetAsync(agg1, 0, ((size_t)ND1 * FEATS + ND1) * sizeof(float), stream);
  sage_scatter<<<(E1 + wpb - 1) / wpb, blk, 0, stream>>>(x, src1, dst1, agg1, deg1, E1);
  sage_gemm<<<ND1 / 16, blk, shmem, stream>>>(x, agg1, deg1, Wself1, Wneigh1, b1,
                                              h1, ND1, /*relu=*/1);
  // ---- Layer 2 ----
  hipMemsetAsync(agg2, 0, ((size_t)ND2 * FEATS + ND2) * sizeof(float), stream);
  sage_scatter<<<(E2 + wpb - 1) / wpb, blk, 0, stream>>>(h1, src2, dst2, agg2, deg2, E2);
  sage_gemm<<<ND2 / 16, blk, shmem, stream>>>(h1, agg2, deg2, Wself2, Wneigh2, b2,
                                              (float*)d_out, ND2, /*relu=*/0);
}